// HGPool_41987600286097
// MI455X (gfx1250) — compile-verified
//
#include <hip/hip_runtime.h>

// Problem constants (match the reference).
#define NB   32
#define NN   1024
#define TOPN 512

// GEMM tiling.
#define TM 32            // rows per block
#define TN 64            // cols per block
#define JB 32            // contraction chunk
#define KTILES (NN / TN) // 16 column-tiles per row

typedef __attribute__((ext_vector_type(2))) float v2f;
typedef __attribute__((ext_vector_type(8))) float v8f;

// ---------------------------------------------------------------------------
// Stage 1: d_inv[b][j] = 1 / sum_i A[b][i][j]   (column sums)
// ---------------------------------------------------------------------------
__global__ void k_colsum_dinv(const float* __restrict__ A,
                              float* __restrict__ dinv) {
    const int b = blockIdx.y;
    const int j = blockIdx.x * blockDim.x + threadIdx.x;
    const float* Ab = A + (size_t)b * NN * NN;
    float s = 0.0f;
#pragma unroll 8
    for (int i = 0; i < NN; ++i)
        s += Ab[(size_t)i * NN + j];
    dinv[b * NN + j] = 1.0f / s;
}

// ---------------------------------------------------------------------------
// Stage 2: per-block fp32 WMMA GEMM tile of DA @ X, fused with
//          residual |X - DA@X| and per-row partial L1 reduction.
//          pPart[b][i][kx] = sum over this block's 64 columns.
// ---------------------------------------------------------------------------
__global__ void __launch_bounds__(256)
k_gemm_l1(const float* __restrict__ A, const float* __restrict__ X,
          const float* __restrict__ dinv, float* __restrict__ pPart) {
    __shared__ float sA[JB][TM + 1];  // j-major (+1 pad: conflict-free stores)
    __shared__ float sX[JB][TN];      // j-major
    __shared__ float pBlk[TM][4];     // per-row partials from the 4 nT waves

    const int b  = blockIdx.z;
    const int iy = blockIdx.y;   // row tile
    const int kx = blockIdx.x;   // col tile

    const int tid  = threadIdx.x;
    const int lane = tid & 31;
    const int wave = tid >> 5;   // 0..7
    const int mT   = wave >> 2;  // 0..1 : 16-row sub-tile
    const int nT   = wave & 3;   // 0..3 : 16-col sub-tile
    const int half = lane >> 4;  // 0/1 : K-half per WMMA layout
    const int l16  = lane & 15;

    const size_t base = (size_t)b * NN * NN;
    const float* Ab = A + base;
    const float* Xb = X + base;

    const int rowBase = iy * TM;
    const int colBase = kx * TN;

    // A-tile loader: tid -> (jj = tid&31, 4 rows iig*4+q). Coalesced along j.
    const int a_jj  = tid & 31;
    const int a_iig = tid >> 5;
    float dscale[4];
#pragma unroll
    for (int q = 0; q < 4; ++q)
        dscale[q] = dinv[b * NN + rowBase + a_iig * 4 + q];

    // X-tile loader: tid -> (col = tid&63, 8 j's). Coalesced along k.
    const int x_col = tid & 63;
    const int x_jg  = tid >> 6;

    v8f acc = {};  // 16x16 f32 accumulator (8 VGPRs, wave32)

    for (int j0 = 0; j0 < NN; j0 += JB) {
        __syncthreads();
#pragma unroll
        for (int q = 0; q < 4; ++q) {
            const int ii = a_iig * 4 + q;
            sA[a_jj][ii] =
                dscale[q] * Ab[(size_t)(rowBase + ii) * NN + (j0 + a_jj)];
        }
#pragma unroll
        for (int q = 0; q < 8; ++q) {
            const int jj = x_jg * 8 + q;
            sX[jj][x_col] = Xb[(size_t)(j0 + jj) * NN + (colBase + x_col)];
        }
        if (j0 + JB < NN) {  // speculative L2 prefetch of next chunk
            __builtin_prefetch(
                &Ab[(size_t)(rowBase + a_iig * 4) * NN + (j0 + JB + a_jj)], 0, 0);
            __builtin_prefetch(
                &Xb[(size_t)(j0 + JB + x_jg * 8) * NN + (colBase + x_col)], 0, 0);
        }
        __syncthreads();

#pragma unroll
        for (int j2 = 0; j2 < JB; j2 += 4) {
            // A 16x4: lane%16 = M, K = half*2 + vgpr
            v2f av, bv;
            av.x = sA[j2 + half * 2 + 0][mT * 16 + l16];
            av.y = sA[j2 + half * 2 + 1][mT * 16 + l16];
            // B 4x16: lane%16 = N, K = half*2 + vgpr
            bv.x = sX[j2 + half * 2 + 0][nT * 16 + l16];
            bv.y = sX[j2 + half * 2 + 1][nT * 16 + l16];
            acc = __builtin_amdgcn_wmma_f32_16x16x4_f32(
                false, av, false, bv, (short)0, acc, false, false);
        }
    }

    // Residual + L1: D layout: vgpr r -> (M = half*8 + r, N = l16).
    const int gk = colBase + nT * 16 + l16;
#pragma unroll
    for (int r = 0; r < 8; ++r) {
        const int rloc = mT * 16 + half * 8 + r;
        const int gi   = rowBase + rloc;
        float m = Xb[(size_t)gi * NN + gk] - acc[r];
        float v = fabsf(m);
        // sum across the 16 lanes of this half (xor < 16 stays in-half)
        v += __shfl_xor(v, 1);
        v += __shfl_xor(v, 2);
        v += __shfl_xor(v, 4);
        v += __shfl_xor(v, 8);
        if (l16 == 0) pBlk[rloc][nT] = v;
    }
    __syncthreads();
    if (tid < TM) {
        // fixed-order reduce over the 4 column sub-tiles -> deterministic
        const float s = pBlk[tid][0] + pBlk[tid][1] + pBlk[tid][2] + pBlk[tid][3];
        pPart[((size_t)b * NN + rowBase + tid) * KTILES + kx] = s;
    }
}

// ---------------------------------------------------------------------------
// Stage 2b: p[b][i] = sum_kx pPart[b][i][kx]  (fixed order, deterministic)
// ---------------------------------------------------------------------------
__global__ void k_reduce_p(const float* __restrict__ pPart,
                           float* __restrict__ p) {
    const int b = blockIdx.y;
    const int i = blockIdx.x * blockDim.x + threadIdx.x;
    const float* src = pPart + ((size_t)b * NN + i) * KTILES;
    float s = 0.0f;
#pragma unroll
    for (int k = 0; k < KTILES; ++k) s += src[k];
    p[b * NN + i] = s;
}

// ---------------------------------------------------------------------------
// Stage 3: per-batch bitonic argsort of (p, index), ascending, lexicographic
//          tie-break on index == stable argsort. Keep first TOPN indices.
// ---------------------------------------------------------------------------
__global__ void __launch_bounds__(NN)
k_sort(const float* __restrict__ p, int* __restrict__ idx) {
    __shared__ float key[NN];
    __shared__ int   val[NN];
    const int b = blockIdx.x;
    const int t = threadIdx.x;
    key[t] = p[b * NN + t];
    val[t] = t;
    __syncthreads();
    for (int size = 2; size <= NN; size <<= 1) {
        for (int stride = size >> 1; stride > 0; stride >>= 1) {
            const int partner = t ^ stride;
            if (partner > t) {
                const bool  up = ((t & size) == 0);
                const float ka = key[t], kb = key[partner];
                const int   va = val[t], vb = val[partner];
                const bool swapAsc  = (kb < ka) || (kb == ka && vb < va);
                const bool swapDesc = (ka < kb) || (ka == kb && va < vb);
                if (up ? swapAsc : swapDesc) {
                    key[t] = kb; key[partner] = ka;
                    val[t] = vb; val[partner] = va;
                }
            }
            __syncthreads();
        }
    }
    if (t < TOPN) idx[b * NN + t] = val[t];
}

// ---------------------------------------------------------------------------
// Stage 4: gathers.
// ---------------------------------------------------------------------------
__global__ void k_gather_A(const float* __restrict__ A,
                           const int* __restrict__ idx,
                           float* __restrict__ An) {
    const int b = blockIdx.z;
    const int r = blockIdx.y;
    const int c = blockIdx.x * blockDim.x + threadIdx.x;
    const int row = idx[b * NN + r];
    const int col = idx[b * NN + c];
    An[((size_t)b * TOPN + r) * TOPN + c] =
        A[((size_t)b * NN + row) * NN + col];
}

__global__ void k_gather_H(const float* __restrict__ X,
                           const int* __restrict__ idx,
                           float* __restrict__ H) {
    const int b = blockIdx.z;
    const int r = blockIdx.y;
    const int k = blockIdx.x * blockDim.x + threadIdx.x;
    const int row = idx[b * NN + r];
    H[((size_t)b * TOPN + r) * NN + k] =
        X[((size_t)b * NN + row) * NN + k];
}

// ---------------------------------------------------------------------------
extern "C" void kernel_launch(void* const* d_in, const int* in_sizes, int n_in,
                              void* d_out, int out_size, void* d_ws,
                              size_t ws_size, hipStream_t stream) {
    const float* A = (const float*)d_in[0];
    const float* X = (const float*)d_in[1];
    // d_in[2] is top_n (compile-time TOPN here)

    float* ws    = (float*)d_ws;
    float* dinv  = ws;                                   // NB*NN
    float* pPart = dinv + (size_t)NB * NN;               // NB*NN*KTILES
    float* p     = pPart + (size_t)NB * NN * KTILES;     // NB*NN
    int*   idx   = (int*)(p + (size_t)NB * NN);          // NB*NN

    float* outA = (float*)d_out;                         // NB*TOPN*TOPN
    float* outH = outA + (size_t)NB * TOPN * TOPN;       // NB*TOPN*NN

    k_colsum_dinv<<<dim3(NN / 256, NB), 256, 0, stream>>>(A, dinv);
    k_gemm_l1<<<dim3(NN / TN, NN / TM, NB), 256, 0, stream>>>(A, X, dinv, pPart);
    k_reduce_p<<<dim3(NN / 256, NB), 256, 0, stream>>>(pPart, p);
    k_sort<<<NB, NN, 0, stream>>>(p, idx);
    k_gather_A<<<dim3(TOPN / 256, TOPN, NB), 256, 0, stream>>>(A, idx, outA);
    k_gather_H<<<dim3(NN / 256, TOPN, NB), 256, 0, stream>>>(X, idx, outH);
}